// ODERGRU_39402029974133
// MI455X (gfx1250) — compile-verified
//
#include <hip/hip_runtime.h>
#include <hip/hip_bf16.h>
#include <math.h>

typedef float v2f __attribute__((ext_vector_type(2)));
typedef float v8f __attribute__((ext_vector_type(8)));

__device__ __forceinline__ v8f wmma_f32_4(v2f a, v2f b, v8f c) {
  // D = A(16x4,f32) * B(4x16,f32) + C(16x16,f32)
  return __builtin_amdgcn_wmma_f32_16x16x4_f32(
      /*neg_a=*/false, a, /*neg_b=*/false, b,
      /*c_mod=*/(short)0, c, /*reuse_a=*/false, /*reuse_b=*/false);
}

__device__ __forceinline__ float fast_tanhf(float x) {
#if __has_builtin(__builtin_amdgcn_tanhf)
  return __builtin_amdgcn_tanhf(x);
#elif __has_builtin(__builtin_amdgcn_tanh_f32)
  return __builtin_amdgcn_tanh_f32(x);
#else
  return tanhf(x);
#endif
}
__device__ __forceinline__ float sigmoidf_(float x) { return 1.f / (1.f + __expf(-x)); }
__device__ __forceinline__ float softplusf_(float x) { return (x > 20.f) ? x : log1pf(__expf(x)); }
__device__ __forceinline__ float bn_lrelu(float v, float sc, float sh) {
  v = fmaf(v, sc, sh);
  return (v > 0.f) ? v : 0.01f * v;
}

// ---------------------------------------------------------------------------
// Fused implicit-GEMM conv1d. BN(+leaky-relu) of the INPUT is folded to
// x*scale+shift (scale/shift staged in LDS). One wave = one 16x16 tile of
// out[Cout x (Nsamp*Tout)] via V_WMMA_F32_16X16X4_F32, K = CIN*KSZ
// (compile-time -> fully unrolled, division-free, branch-free inner loop).
// grid: (Ntiles/8, Cout/16), block 256 (8 waves).
// ---------------------------------------------------------------------------
template <int CIN, int KSZ, int PAD, bool DOBN>
__global__ __launch_bounds__(256) void conv_bn_wmma(
    const float* __restrict__ in, const float* __restrict__ W,
    const float* __restrict__ bias,
    const float* __restrict__ psc, const float* __restrict__ psh,
    float* __restrict__ out, float* __restrict__ sSum, float* __restrict__ sSq,
    int Tin, int Tout, int Cout)
{
  static_assert(KSZ == 2 ? PAD == 0 : true, "k2 convs are pad-0 here");
  constexpr int Kg = CIN * KSZ;
  __shared__ float lsum[16], lsq[16];
  __shared__ float sc[CIN], sh[CIN];
  const int tid = threadIdx.x;
  if (tid < 16) { lsum[tid] = 0.f; lsq[tid] = 0.f; }
  if (DOBN) {
    for (int i = tid; i < CIN; i += 256) { sc[i] = psc[i]; sh[i] = psh[i]; }
  }
  __syncthreads();

  const int lane = tid & 31, wave = tid >> 5;
  const int half = lane >> 4, li = lane & 15;
  const int tileN = blockIdx.x * 8 + wave;
  const int mBase = blockIdx.y * 16;
  const int nIdx = tileN * 16 + li;          // B column = (sample, t)
  const int samp = nIdx / Tout;
  const int t = nIdx - samp * Tout;
  const float* inS = in + (size_t)samp * CIN * Tin;
  const int am = mBase + li;                 // A row = output channel
  const float* Wr = W + (size_t)am * Kg;

  v8f c = {};
  if constexpr (KSZ == 2) {
    // pad==0, kernel 2: columns (t, t+1) always in bounds (t <= Tin-2)
#pragma unroll
    for (int k0 = 0; k0 < Kg; k0 += 4) {
      const int kb = k0 + 2 * half;
      const int ci = kb >> 1;                // kb even -> both taps same channel
      v2f a, b;
      a.x = Wr[kb]; a.y = Wr[kb + 1];
      float v0 = inS[ci * Tin + t];
      float v1 = inS[ci * Tin + t + 1];
      if constexpr (DOBN) {
        const float s_ = sc[ci], h_ = sh[ci];
        v0 = bn_lrelu(v0, s_, h_);
        v1 = bn_lrelu(v1, s_, h_);
      }
      b.x = v0; b.y = v1;
      c = wmma_f32_4(a, b, c);
    }
  } else {
    // kernel 1 with padding: in-bounds test depends only on t -> hoisted
    const int ti = t - PAD;
    const bool ok = (ti >= 0) && (ti < Tin);
#pragma unroll
    for (int k0 = 0; k0 < Kg; k0 += 4) {
      const int kb = k0 + 2 * half;          // ci == kb for KSZ==1
      v2f a, b;
      a.x = Wr[kb]; a.y = Wr[kb + 1];
      float v0 = 0.f, v1 = 0.f;
      if (ok) {
        v0 = inS[kb * Tin + ti];
        v1 = inS[(kb + 1) * Tin + ti];
        if constexpr (DOBN) {
          v0 = bn_lrelu(v0, sc[kb], sh[kb]);
          v1 = bn_lrelu(v1, sc[kb + 1], sh[kb + 1]);
        }
      }
      b.x = v0; b.y = v1;
      c = wmma_f32_4(a, b, c);
    }
  }

  // store conv+bias (raw pre-BN); accumulate per-channel stats for this layer
  float partS[8], partQ[8];
  const size_t outS = (size_t)samp * Cout * Tout;
#pragma unroll
  for (int r = 0; r < 8; ++r) {
    const int m = mBase + r + 8 * half;
    const float v = c[r] + bias[m];
    out[outS + (size_t)m * Tout + t] = v;
    partS[r] = v; partQ[r] = v * v;
  }

  if (sSum) {
#pragma unroll
    for (int r = 0; r < 8; ++r) {
      for (int mask = 1; mask <= 8; mask <<= 1) {
        partS[r] += __shfl_xor(partS[r], mask, 32);
        partQ[r] += __shfl_xor(partQ[r], mask, 32);
      }
    }
    if (li == 0) {
#pragma unroll
      for (int r = 0; r < 8; ++r) {
        atomicAdd(&lsum[r + 8 * half], partS[r]);
        atomicAdd(&lsq [r + 8 * half], partQ[r]);
      }
    }
    __syncthreads();
    if (tid < 16) {
      atomicAdd(&sSum[mBase + tid], lsum[tid]);
      atomicAdd(&sSq [mBase + tid], lsq[tid]);
    }
  }
}

// BN fold: (x-m)*rsqrt(v+eps)*g + b  ==  x*scale + shift
__global__ void bn_finalize(const float* __restrict__ s, const float* __restrict__ q,
                            const float* __restrict__ g, const float* __restrict__ bb,
                            float* __restrict__ scale, float* __restrict__ shift,
                            int C, float count)
{
  const int c = threadIdx.x;
  if (c < C) {
    const float m = s[c] / count;
    const float v = q[c] / count - m * m;
    const float sc = rsqrtf(v + 1e-5f) * g[c];
    scale[c] = sc;
    shift[c] = bb[c] - m * sc;
  }
}

// ---------------------------------------------------------------------------
// Per-sample OAS-shrunk covariance + Cholesky. One wave32 per sample.
// z: (4096, 32, 63) -> X (63 x 32), emp = Xc^T Xc / 62 via WMMA f32 16x16x4.
// ---------------------------------------------------------------------------
__global__ __launch_bounds__(32) void cov_oas_chol(
    const float* __restrict__ z, float* __restrict__ xd, float* __restrict__ xl)
{
  __shared__ float Xc[64][32];   // K padded 63 -> 64 with a zero row
  __shared__ float Cv[32][33];
  const int lane = threadIdx.x;
  const int samp = blockIdx.x;
  const float* zs = z + (size_t)samp * 32 * 63;

  float s = 0.f;
  for (int m = 0; m < 63; ++m) {
    const float v = zs[lane * 63 + m];
    Xc[m][lane] = v;
    s += v;
  }
  const float mean = s * (1.f / 63.f);
  for (int m = 0; m < 63; ++m) Xc[m][lane] -= mean;
  Xc[63][lane] = 0.f;
  __syncthreads();

  const int half = lane >> 4, li = lane & 15;
#pragma unroll
  for (int mt = 0; mt < 2; ++mt)
#pragma unroll
    for (int nt = 0; nt < 2; ++nt) {
      v8f c = {};
#pragma unroll 8
      for (int k0 = 0; k0 < 64; k0 += 4) {
        const int kb = k0 + 2 * half;
        v2f a, b;
        a.x = Xc[kb][mt * 16 + li]; a.y = Xc[kb + 1][mt * 16 + li];
        b.x = Xc[kb][nt * 16 + li]; b.y = Xc[kb + 1][nt * 16 + li];
        c = wmma_f32_4(a, b, c);
      }
      const int n = nt * 16 + li;
#pragma unroll
      for (int r = 0; r < 8; ++r)
        Cv[mt * 16 + r + 8 * half][n] = c[r] * (1.f / 62.f);
    }
  __syncthreads();

  // OAS shrinkage
  float diag = Cv[lane][lane];
  float sq = 0.f;
  for (int j = 0; j < 32; ++j) { const float e = Cv[lane][j]; sq += e * e; }
  for (int mask = 1; mask < 32; mask <<= 1) {
    diag += __shfl_xor(diag, mask, 32);
    sq   += __shfl_xor(sq,   mask, 32);
  }
  const float mu = diag * (1.f / 32.f);
  const float alpha = sq * (1.f / 1024.f);
  const float num = alpha + mu * mu;
  const float den = 64.f * (alpha - mu * mu * (1.f / 32.f));
  const float shr = (den == 0.f) ? 1.f : fminf(num / den, 1.f);
  for (int j = 0; j < 32; ++j) {
    float v = (1.f - shr) * Cv[lane][j];
    if (j == lane) v += shr * mu;
    Cv[lane][j] = v;
  }
  __syncthreads();

  // Cholesky (row-parallel, lane i owns row i)
  for (int j = 0; j < 32; ++j) {
    float sacc = 0.f;
    for (int k = 0; k < j; ++k) sacc += Cv[lane][k] * Cv[j][k];
    if (lane == j) Cv[j][j] = sqrtf(Cv[j][j] - sacc);
    __syncthreads();
    if (lane > j) Cv[lane][j] = (Cv[lane][j] - sacc) / Cv[j][j];
    __syncthreads();
  }

  xd[(size_t)samp * 32 + lane] = Cv[lane][lane];
  for (int idx = lane; idx < 496; idx += 32) {   // np.tril_indices(32,-1) order
    int r = (int)((1.f + sqrtf(1.f + 8.f * (float)idx)) * 0.5f);
    while (r * (r - 1) / 2 > idx) --r;
    while ((r + 1) * r / 2 <= idx) ++r;
    const int cc = idx - r * (r - 1) / 2;
    xl[(size_t)samp * 496 + idx] = Cv[r][cc];
  }
}

// ---------------------------------------------------------------------------
// Scan-phase tiled GEMM: Out(32xN) = A(32xK, LDS) * op(W, global) [+ bias]
// WMODE 0: W is (N,K) row-major (x @ W^T);  1: W is (K,N);  2: (K,N) with abs
// N, K compile-time -> unrolled, branch-free inner loop.
// ---------------------------------------------------------------------------
template <int WMODE, int N, int K>
__device__ __forceinline__ void gemm_tile(
    const float* __restrict__ A, int lda, const float* __restrict__ W,
    const float* __restrict__ bias, float* __restrict__ Out, int ldo,
    int nWaves, int wave, int lane)
{
  constexpr int ntiles = (N + 15) >> 4;
  constexpr int total = 2 * ntiles;          // M = 32 -> 2 M-tiles
  const int half = lane >> 4, li = lane & 15;
  for (int tile = wave; tile < total; tile += nWaves) {
    const int mt = tile & 1, nt = tile >> 1;
    const int am = mt * 16 + li;
    const int bn = nt * 16 + li;
    const bool nok = bn < N;
    const float* Ar = A + am * lda;
    v8f c = {};
#pragma unroll 4
    for (int k0 = 0; k0 < K; k0 += 4) {
      const int kb = k0 + 2 * half;
      v2f a, b;
      a.x = Ar[kb]; a.y = Ar[kb + 1];
      if constexpr (WMODE == 0) {
        b.x = nok ? W[(size_t)bn * K + kb] : 0.f;
        b.y = nok ? W[(size_t)bn * K + kb + 1] : 0.f;
      } else {
        float w0 = nok ? W[(size_t)kb * N + bn] : 0.f;
        float w1 = nok ? W[(size_t)(kb + 1) * N + bn] : 0.f;
        if constexpr (WMODE == 2) { w0 = fabsf(w0); w1 = fabsf(w1); }
        b.x = w0; b.y = w1;
      }
      c = wmma_f32_4(a, b, c);
    }
    if (nok) {
      const float bb = bias ? bias[bn] : 0.f;
#pragma unroll
      for (int r = 0; r < 8; ++r) {
        const int m = mt * 16 + r + 8 * half;
        Out[m * ldo + bn] = c[r] + bb;
      }
    }
  }
}

// ---------------------------------------------------------------------------
// Persistent single-workgroup ODE-RGRU scan: 128 sequential steps, batch 32.
// All recurrent state + staged per-step inputs live in ~268KB dynamic LDS
// (WGP has 320KB). Weights stream from the 192MB L2. 1024 threads = 32 waves.
// ---------------------------------------------------------------------------
__global__ __launch_bounds__(1024) void oder_scan(
    const float* __restrict__ xd, const float* __restrict__ xl,
    const float* __restrict__ wi, const float* __restrict__ bi,
    const float* __restrict__ w0, const float* __restrict__ b0,
    const float* __restrict__ w1, const float* __restrict__ b1,
    const float* __restrict__ wo, const float* __restrict__ bo,
    const float* __restrict__ gdwx, const float* __restrict__ gdwh, const float* __restrict__ gdb,
    const float* __restrict__ glwx, const float* __restrict__ glwh, const float* __restrict__ glb,
    const float* __restrict__ clsw, const float* __restrict__ clsb,
    float* __restrict__ outp)
{
  extern __shared__ float sm[];
  float* y   = sm;               // 32*136
  float* ht1 = y   + 32 * 136;   // 32*128
  float* ht2 = ht1 + 32 * 128;   // 32*128
  float* ob  = ht2 + 32 * 128;   // 32*136
  float* hd  = ob  + 32 * 136;   // 32*16
  float* hl  = hd  + 32 * 16;    // 32*120
  float* gxd = hl  + 32 * 120;   // 32*48
  float* ghd = gxd + 32 * 48;    // 32*48
  float* xdt = ghd + 32 * 48;    // 32*32
  float* xlt = xdt + 32 * 32;    // 32*496
  float* gbx = xlt + 32 * 496;   // 32*360
  float* gbh = gbx + 32 * 360;   // 32*360
  float* acc = gbh + 32 * 360;   // 32*136

  const int tid = threadIdx.x, nthr = blockDim.x;
  const int lane = tid & 31, wave = tid >> 5, nW = nthr >> 5;

  for (int i = tid; i < 32 * 16;  i += nthr) hd[i] = 1.f;
  for (int i = tid; i < 32 * 120; i += nthr) hl[i] = 0.f;
  for (int i = tid; i < 32 * 136; i += nthr) acc[i] = 0.f;
  __syncthreads();

  for (int step = 0; step < 128; ++step) {
    // y = [log(hd), hl]; stage xd_t, xl_t (time-major slices of (b,s,*))
    for (int i = tid; i < 32 * 136; i += nthr) {
      const int b = i / 136, j = i - b * 136;
      y[i] = (j < 16) ? __logf(hd[b * 16 + j]) : hl[b * 120 + (j - 16)];
    }
    for (int i = tid; i < 32 * 32; i += nthr) {
      const int b = i >> 5, f = i & 31;
      xdt[i] = xd[((size_t)b * 128 + step) * 32 + f];
    }
    for (int i = tid; i < 32 * 496; i += nthr) {
      const int b = i / 496, f = i - b * 496;
      xlt[i] = xl[((size_t)b * 128 + step) * 496 + f];
    }
    if (step + 1 < 128) {  // pull next step's gather lines toward the WGP
      for (int i = tid; i < 32; i += nthr)
        __builtin_prefetch(&xl[((size_t)i * 128 + step + 1) * 496], 0, 0);
    }
    __syncthreads();

    // odef: 4 GEMMs with tanh between
    gemm_tile<0, 128, 136>(y, 136, wi, bi, ht1, 128, nW, wave, lane);
    __syncthreads();
    for (int i = tid; i < 32 * 128; i += nthr) ht1[i] = fast_tanhf(ht1[i]);
    __syncthreads();
    gemm_tile<0, 128, 128>(ht1, 128, w0, b0, ht2, 128, nW, wave, lane);
    __syncthreads();
    for (int i = tid; i < 32 * 128; i += nthr) ht2[i] = fast_tanhf(ht2[i]);
    __syncthreads();
    gemm_tile<0, 128, 128>(ht2, 128, w1, b1, ht1, 128, nW, wave, lane);
    __syncthreads();
    for (int i = tid; i < 32 * 128; i += nthr) ht1[i] = fast_tanhf(ht1[i]);
    __syncthreads();
    gemm_tile<0, 136, 128>(ht1, 128, wo, bo, ob, 136, nW, wave, lane);
    __syncthreads();

    // hp = y - odef(y); hd_ode = exp(tanh(hp[:, :16])); hl_ode = hp[:, 16:]
    for (int i = tid; i < 32 * 136; i += nthr) {
      const int b = i / 136, j = i - b * 136;
      const float hp = y[i] - ob[i];
      if (j < 16) hd[b * 16 + j] = __expf(fast_tanhf(hp));
      else        hl[b * 120 + (j - 16)] = hp;
    }
    __syncthreads();

    // GRU gate GEMMs (independent outputs -> no barriers between them)
    gemm_tile<2, 48, 32>(xdt, 32, gdwx, nullptr, gxd, 48, nW, wave, lane);
    gemm_tile<2, 48, 16>(hd, 16, gdwh, nullptr, ghd, 48, nW, wave, lane);
    gemm_tile<0, 360, 496>(xlt, 496, glwx, nullptr, gbx, 360, nW, wave, lane);
    gemm_tile<0, 360, 120>(hl, 120, glwh, nullptr, gbh, 360, nW, wave, lane);
    __syncthreads();

    // gru_d (log-domain GRU on positive state)
    for (int i = tid; i < 32 * 16; i += nthr) {
      const int b = i >> 4, u = i & 15;
      const float ir = gxd[b * 48 + u], ii = gxd[b * 48 + 16 + u], in_ = gxd[b * 48 + 32 + u];
      const float hr = ghd[b * 48 + u], hi = ghd[b * 48 + 16 + u], hn  = ghd[b * 48 + 32 + u];
      const float br = fabsf(gdb[u]), bz = fabsf(gdb[16 + u]), bn = fabsf(gdb[32 + u]);
      const float r  = sigmoidf_(br * ir * hr);   // sigmoid(br*exp(log ir + log hr))
      const float zg = sigmoidf_(bz * ii * hi);
      const float ng = softplusf_(bn * in_ * (r * hn));
      const float h0 = hd[i];
      hd[i] = __expf((1.f - zg) * __logf(ng) + zg * __logf(h0));
    }
    // gru_l (standard GRU)
    for (int i = tid; i < 32 * 120; i += nthr) {
      const int b = i / 120, u = i - b * 120;
      const float r  = sigmoidf_(gbx[b * 360 + u] + gbh[b * 360 + u] + glb[u]);
      const float zg = sigmoidf_(gbx[b * 360 + 120 + u] + gbh[b * 360 + 120 + u] + glb[120 + u]);
      const float ng = fast_tanhf(gbx[b * 360 + 240 + u] + r * gbh[b * 360 + 240 + u] + glb[240 + u]);
      const float h0 = hl[i];
      hl[i] = ng + zg * (h0 - ng);
    }
    __syncthreads();

    // outs accumulation: acc += [log(hd), hl]
    for (int i = tid; i < 32 * 136; i += nthr) {
      const int b = i / 136, j = i - b * 136;
      acc[i] += (j < 16) ? __logf(hd[b * 16 + j]) : hl[b * 120 + (j - 16)];
    }
    __syncthreads();
  }

  // logits = mean(outs) @ cls_w^T + cls_b  -> (32,10)
  for (int i = tid; i < 320; i += nthr) {
    const int b = i / 10, o = i - b * 10;
    float s = clsb[o];
    for (int j = 0; j < 136; ++j)
      s += (acc[b * 136 + j] * (1.f / 128.f)) * clsw[o * 136 + j];
    outp[i] = s;
  }
}

// ---------------------------------------------------------------------------
extern "C" void kernel_launch(void* const* d_in, const int* in_sizes, int n_in,
                              void* d_out, int out_size, void* d_ws, size_t ws_size,
                              hipStream_t stream) {
  const float* x    = (const float*)d_in[0];
  const float* c0w  = (const float*)d_in[1];  const float* c0b = (const float*)d_in[2];
  const float* bn0g = (const float*)d_in[3];  const float* bn0b = (const float*)d_in[4];
  const float* c1w  = (const float*)d_in[5];  const float* c1b = (const float*)d_in[6];
  const float* bn1g = (const float*)d_in[7];  const float* bn1b = (const float*)d_in[8];
  const float* c2w  = (const float*)d_in[9];  const float* c2b = (const float*)d_in[10];
  const float* bn2g = (const float*)d_in[11]; const float* bn2b = (const float*)d_in[12];
  const float* c3w  = (const float*)d_in[13]; const float* c3b = (const float*)d_in[14];
  const float* wi   = (const float*)d_in[15]; const float* bi  = (const float*)d_in[16];
  const float* w0   = (const float*)d_in[17]; const float* b0  = (const float*)d_in[18];
  const float* w1   = (const float*)d_in[19]; const float* b1  = (const float*)d_in[20];
  const float* wo   = (const float*)d_in[21]; const float* bo  = (const float*)d_in[22];
  const float* gdwx = (const float*)d_in[23]; const float* gdwh = (const float*)d_in[24];
  const float* gdb  = (const float*)d_in[25];
  const float* glwx = (const float*)d_in[26]; const float* glwh = (const float*)d_in[27];
  const float* glb  = (const float*)d_in[28];
  const float* clsw = (const float*)d_in[29]; const float* clsb = (const float*)d_in[30];
  float* out = (float*)d_out;

  float* ws = (float*)d_ws;
  // stats region (zeroed each call): sum/sumsq + folded scale/shift per layer
  float* s0 = ws +   0; float* q0 = ws +  64;
  float* s1 = ws + 128; float* q1 = ws + 192;
  float* s2 = ws + 256; float* q2 = ws + 320;
  float* sc0 = ws + 384; float* sh0 = ws + 448;
  float* sc1 = ws + 512; float* sh1 = ws + 576;
  float* sc2 = ws + 640; float* sh2 = ws + 704;

  const size_t NS = 4096;
  float* z0 = ws + 1024;                          // 4096*32*63
  float* z1 = z0 + NS * 32 * 63;                  // 4096*48*62
  float* z2 = z1 + NS * 48 * 62;                  // 4096*64*61
  float* z3 = z0;                                 // reuse (z0 dead after conv1)
  float* xd = z2 + NS * 64 * 61;                  // 4096*32
  float* xl = xd + NS * 32;                       // 4096*496

  hipMemsetAsync(ws, 0, 1024 * sizeof(float), stream);

  // conv0: 64->32, k2, T 64->63. Ntiles = 4096*63/16 = 16128 -> grid.x 2016
  conv_bn_wmma<64, 2, 0, false><<<dim3(2016, 2), 256, 0, stream>>>(
      x, c0w, c0b, nullptr, nullptr, z0, s0, q0, 64, 63, 32);
  bn_finalize<<<1, 64, 0, stream>>>(s0, q0, bn0g, bn0b, sc0, sh0, 32, (float)(NS * 63));

  // conv1: 32->48, k2, T 63->62. Ntiles = 15872 -> 1984
  conv_bn_wmma<32, 2, 0, true><<<dim3(1984, 3), 256, 0, stream>>>(
      z0, c1w, c1b, sc0, sh0, z1, s1, q1, 63, 62, 48);
  bn_finalize<<<1, 64, 0, stream>>>(s1, q1, bn1g, bn1b, sc1, sh1, 48, (float)(NS * 62));

  // conv2: 48->64, k2, T 62->61. Ntiles = 15616 -> 1952
  conv_bn_wmma<48, 2, 0, true><<<dim3(1952, 4), 256, 0, stream>>>(
      z1, c2w, c2b, sc1, sh1, z2, s2, q2, 62, 61, 64);
  bn_finalize<<<1, 64, 0, stream>>>(s2, q2, bn2g, bn2b, sc2, sh2, 64, (float)(NS * 61));

  // conv3: 64->32, k1 pad1, T 61->63 (no BN after). Ntiles = 16128 -> 2016
  conv_bn_wmma<64, 1, 1, true><<<dim3(2016, 2), 256, 0, stream>>>(
      z2, c3w, c3b, sc2, sh2, z3, nullptr, nullptr, 61, 63, 32);

  // per-sample OAS covariance + Cholesky
  cov_oas_chol<<<4096, 32, 0, stream>>>(z3, xd, xl);

  // persistent scan: one WG, 32 waves, ~268KB dynamic LDS
  const size_t smemFloats =
      32 * (136 + 128 + 128 + 136 + 16 + 120 + 48 + 48 + 32 + 496 + 360 + 360 + 136);
  oder_scan<<<1, 1024, smemFloats * sizeof(float), stream>>>(
      xd, xl, wi, bi, w0, b0, w1, b1, wo, bo,
      gdwx, gdwh, gdb, glwx, glwh, glb, clsw, clsb, out);
}